// Quanv3x3LayerClass_32942399161028
// MI455X (gfx1250) — compile-verified
//
#include <hip/hip_runtime.h>

typedef __attribute__((ext_vector_type(2))) float v2f;
typedef __attribute__((ext_vector_type(8))) float v8f;

#define PI_F 3.14159265358979323846f

// ---------------------------------------------------------------------------
// Qubit -> state-bit mapping (chosen to match WMMA operand layouts):
//   r-bits (M dim, 16):  q0->r3, q1->r2, q2->r1, q3->r0      (r = lane & 15)
//   c-bits (K/N dim,32): q4->c4, q5->c3, q6->c2, q7->c1, q8->c0
// A-operand storage (state, 16 complex per lane):
//   value i (0..15) holds element (M = lane&15, c = 4*(i>>1) + (i&1) + 2*(lane>>4))
//   -> chunk kc uses {val[2kc], val[2kc+1]} as the 2-VGPR A operand.
// C/D layout: VGPR j: lanes 0-15 -> M=j, lanes 16-31 -> M=j+8, N = lane&15.
// B layout (assumed, mirrors A): VGPR0: K=0 (lanes<16) / K=2 (lanes>=16),
//   VGPR1: K=1 / K=3, N = lane&15.
// ---------------------------------------------------------------------------

__global__ void zero_out_kernel(float* __restrict__ out, int n) {
  int i = blockIdx.x * blockDim.x + threadIdx.x;
  if (i < n) out[i] = 0.0f;
}

// Precompute L_k = gates 4..7 (CRZ+CRX pairs) on qubits {4..8} as a 32x32
// complex matrix, stored in WMMA-B layout. One wave per filter k.
// Gate pairs on c-bits: (t6,t7): ctrl c0 tgt c1 ; (t8,t9): ctrl c3 tgt c4 ;
//                       (t10,t11): ctrl c1 tgt c2 ; (t12,t13): ctrl c2 tgt c4
__global__ __launch_bounds__(32) void precompute_L_kernel(
    const float* __restrict__ qp, float* __restrict__ Lb) {
  const int k = blockIdx.x;
  const int lane = threadIdx.x;              // = output row c of L
  const float* t = qp + k * 16;

  float xre[32], xim[32];                    // X[c=lane][c'=j], starts identity
#pragma unroll
  for (int j = 0; j < 32; ++j) { xre[j] = (j == lane) ? 1.f : 0.f; xim[j] = 0.f; }

  const int G[4][3] = {{6,0,1},{8,3,4},{10,1,2},{12,2,4}}; // {tb, ctrlbit, tgtbit}
#pragma unroll
  for (int g = 0; g < 4; ++g) {
    const int cb = G[g][1], tb = G[g][2];
    const int ctrl = (lane >> cb) & 1;
    const int tgt  = (lane >> tb) & 1;
    { // CRZ(theta = t[tb]) : control=1 -> phase exp(+/- i th/2) by target bit
      float h = 0.5f * t[G[g][0]];
      float cz = cosf(h), sz = sinf(h);
      float cc = ctrl ? cz : 1.0f;
      float gg = ctrl ? (tgt ? sz : -sz) : 0.0f;
#pragma unroll
      for (int j = 0; j < 32; ++j) {
        float re = xre[j], im = xim[j];
        xre[j] = re * cc - im * gg;
        xim[j] = im * cc + re * gg;
      }
    }
    { // CRX(theta = t[tb+1]) : new = co*a + (-i si)*partner   (ctrl=1 lanes)
      float h = 0.5f * t[G[g][0] + 1];
      float co = cosf(h), si = sinf(h);
      float ce = ctrl ? co : 1.0f;
      float se = ctrl ? si : 0.0f;
#pragma unroll
      for (int j = 0; j < 32; ++j) {
        float pre = __shfl_xor(xre[j], 1 << tb, 32);
        float pim = __shfl_xor(xim[j], 1 << tb, 32);
        float re = xre[j], im = xim[j];
        xre[j] = ce * re + se * pim;
        xim[j] = ce * im - se * pre;
      }
    }
  }

  // Store B-layout: per (k, tile tt, chunk kc): 64 floats re + 64 floats im.
  // B[K=4kc+kk][N=l&15] = L[16*tt + N][4kc + kk], kk = v + 2*(lane>>4).
  const int nlow = lane & 15;
  const int hi = lane >> 4;
#pragma unroll
  for (int tt = 0; tt < 2; ++tt) {
#pragma unroll
    for (int kc = 0; kc < 8; ++kc) {
      float* base = Lb + (((k * 16) + tt * 8 + kc) << 7);  // *128 floats
#pragma unroll
      for (int v = 0; v < 2; ++v) {
        float r0 = __shfl(xre[4 * kc + v],     16 * tt + nlow, 32);
        float r2 = __shfl(xre[4 * kc + v + 2], 16 * tt + nlow, 32);
        float i0 = __shfl(xim[4 * kc + v],     16 * tt + nlow, 32);
        float i2 = __shfl(xim[4 * kc + v + 2], 16 * tt + nlow, 32);
        base[2 * lane + v]      = hi ? r2 : r0;
        base[64 + 2 * lane + v] = hi ? i2 : i0;
      }
    }
  }
}

// One wave32 per (patch, filter) circuit. 8 waves / 256-thread block.
__global__ __launch_bounds__(256) void quanv_main_kernel(
    const float* __restrict__ x, const float* __restrict__ qp,
    const float* __restrict__ Lb, float* __restrict__ out) {
  const int lane = threadIdx.x & 31;
  const int wave = (blockIdx.x << 3) + (threadIdx.x >> 5);
  const int k = wave & 31;          // filter
  const int p = wave >> 5;          // patch id 0..967
  const int b = p / 484;
  const int rem = p - b * 484;
  const int row = rem / 22;
  const int col = rem - row * 22;

  // --- per-qubit RX(pi*d) trig ---
  float cq[9], sq[9];
#pragma unroll
  for (int m = 0; m < 9; ++m) {
    float d = x[(b * 24 + row + m / 3) * 24 + (col + m % 3)];
    float h = (0.5f * PI_F) * d;
    cq[m] = cosf(h);
    sq[m] = sinf(h);
  }
  const float* t = qp + k * 16;

  // --- initial product state directly in WMMA A-layout ---
  const int r  = lane & 15;
  const int l4 = lane >> 4;         // = c-bit 1 = qubit 7
  float mlane = ((r >> 3) & 1 ? sq[0] : cq[0]) * ((r >> 2) & 1 ? sq[1] : cq[1]) *
                ((r >> 1) & 1 ? sq[2] : cq[2]) * ((r & 1)      ? sq[3] : cq[3]) *
                (l4 ? sq[7] : cq[7]);
  int nlane = ((r >> 3) & 1) + ((r >> 2) & 1) + ((r >> 1) & 1) + (r & 1) + l4;

  float sre[16], sim[16];
#pragma unroll
  for (int i = 0; i < 16; ++i) {
    // i bits: bit0->q8, bit1->q6, bit2->q5, bit3->q4
    float mi = ((i & 1)        ? sq[8] : cq[8]) *
               (((i >> 1) & 1) ? sq[6] : cq[6]) *
               (((i >> 2) & 1) ? sq[5] : cq[5]) *
               (((i >> 3) & 1) ? sq[4] : cq[4]);
    int n = (nlane + (i & 1) + ((i >> 1) & 1) + ((i >> 2) & 1) + ((i >> 3) & 1)) & 3;
    float m = mlane * mi;           // amplitude = m * (-i)^n
    sre[i] = (n == 0) ? m : ((n == 2) ? -m : 0.f);
    sim[i] = (n == 1) ? -m : ((n == 3) ? m : 0.f);
  }

  // --- gates 1..3 on r-bits (lane bits 0..3), cross-lane butterflies ---
  const int RG[3][3] = {{0,2,3},{2,0,1},{4,1,3}}; // {theta_base, ctrl_lanebit, tgt_lanebit}
#pragma unroll
  for (int g = 0; g < 3; ++g) {
    const int cb = RG[g][1], tb = RG[g][2];
    const int ctrl = (lane >> cb) & 1;
    const int tgt  = (lane >> tb) & 1;
    { // CRZ
      float h = 0.5f * t[RG[g][0]];
      float cz = cosf(h), szv = sinf(h);
      float cc = ctrl ? cz : 1.f;
      float gg = ctrl ? (tgt ? szv : -szv) : 0.f;
#pragma unroll
      for (int i = 0; i < 16; ++i) {
        float re = sre[i], im = sim[i];
        sre[i] = re * cc - im * gg;
        sim[i] = im * cc + re * gg;
      }
    }
    { // CRX
      float h = 0.5f * t[RG[g][0] + 1];
      float co = cosf(h), si = sinf(h);
      float ce = ctrl ? co : 1.f;
      float se = ctrl ? si : 0.f;
#pragma unroll
      for (int i = 0; i < 16; ++i) {
        float pre = __shfl_xor(sre[i], 1 << tb, 32);
        float pim = __shfl_xor(sim[i], 1 << tb, 32);
        float re = sre[i], im = sim[i];
        sre[i] = ce * re + se * pim;
        sim[i] = ce * im - se * pre;
      }
    }
  }

  // --- gates 4..7 as S' = S @ L^T via v_wmma_f32_16x16x4_f32 ---
  float nimv[16];
#pragma unroll
  for (int i = 0; i < 16; ++i) nimv[i] = -sim[i];

  v8f dre[2], dim_[2];
#pragma unroll
  for (int tt = 0; tt < 2; ++tt) {
    v8f accre = {}, accim = {};
#pragma unroll
    for (int kc = 0; kc < 8; ++kc) {
      const v2f* bb = (const v2f*)(Lb + (((k * 16) + tt * 8 + kc) << 7));
      v2f bre = bb[lane];        // B chunk, real
      v2f bim = bb[32 + lane];   // B chunk, imag
      v2f are  = {sre[2 * kc],  sre[2 * kc + 1]};
      v2f aim  = {sim[2 * kc],  sim[2 * kc + 1]};
      v2f anim = {nimv[2 * kc], nimv[2 * kc + 1]};
      // D_re += A_re*B_re - A_im*B_im ; D_im += A_re*B_im + A_im*B_re
      accre = __builtin_amdgcn_wmma_f32_16x16x4_f32(false, are,  false, bre,
                                                    (short)0, accre, false, false);
      accre = __builtin_amdgcn_wmma_f32_16x16x4_f32(false, anim, false, bim,
                                                    (short)0, accre, false, false);
      accim = __builtin_amdgcn_wmma_f32_16x16x4_f32(false, are,  false, bim,
                                                    (short)0, accim, false, false);
      accim = __builtin_amdgcn_wmma_f32_16x16x4_f32(false, aim,  false, bre,
                                                    (short)0, accim, false, false);
    }
    dre[tt] = accre;
    dim_[tt] = accim;
  }

  // --- gate 8: CRZ(t14)+CRX(t15), ctrl = qubit4 = N-tile, tgt = qubit0 = M[3] ---
  // Only tile 1 (ctrl=1) is modified. M[3] == (lane >= 16); partner = lane^16.
  {
    float h = 0.5f * t[14];
    float cz = cosf(h), szv = sinf(h);
    float gg = (lane < 16) ? -szv : szv;
#pragma unroll
    for (int j = 0; j < 8; ++j) {
      float re = dre[1][j], im = dim_[1][j];
      dre[1][j] = re * cz - im * gg;
      dim_[1][j] = im * cz + re * gg;
    }
  }
  {
    float h = 0.5f * t[15];
    float co = cosf(h), si = sinf(h);
#pragma unroll
    for (int j = 0; j < 8; ++j) {
      float pre = __shfl_xor((float)dre[1][j], 16, 32);
      float pim = __shfl_xor((float)dim_[1][j], 16, 32);
      float re = dre[1][j], im = dim_[1][j];
      dre[1][j] = co * re + si * pim;
      dim_[1][j] = co * im - si * pre;
    }
  }

  // --- measurement: qubit0 = M[3]; lanes<16 -> p0, lanes>=16 -> p1 ---
  float s = 0.f;
#pragma unroll
  for (int tt = 0; tt < 2; ++tt)
#pragma unroll
    for (int j = 0; j < 8; ++j)
      s += dre[tt][j] * dre[tt][j] + dim_[tt][j] * dim_[tt][j];
  float v = (lane < 16) ? s : -s;   // contributes p0 - p1
#pragma unroll
  for (int off = 16; off >= 1; off >>= 1) v += __shfl_xor(v, off, 32);

  if (lane == 0)
    out[((b * 24 + row + 1) * 24 + (col + 1)) * 32 + k] = (v + 1.f) * 0.5f;
}

extern "C" void kernel_launch(void* const* d_in, const int* in_sizes, int n_in,
                              void* d_out, int out_size, void* d_ws, size_t ws_size,
                              hipStream_t stream) {
  const float* x  = (const float*)d_in[0];   // (2,24,24,1) float32
  const float* qp = (const float*)d_in[1];   // (32,16) float32
  float* out = (float*)d_out;                // (2,24,24,32) float32
  float* Lb  = (float*)d_ws;                 // 32*16*128 floats = 256 KB

  zero_out_kernel<<<(out_size + 255) / 256, 256, 0, stream>>>(out, out_size);
  precompute_L_kernel<<<32, 32, 0, stream>>>(qp, Lb);
  // 968 patches * 32 filters = 30976 waves; 8 waves/block -> 3872 blocks
  quanv_main_kernel<<<3872, 256, 0, stream>>>(x, qp, Lb, out);
}